// EdgeAligner_66365834658202
// MI455X (gfx1250) — compile-verified
//
#include <hip/hip_runtime.h>
#include <hip/hip_bf16.h>

// EdgeAligner: KNN edge features + fp32-WMMA cross attention + spatial proj.
// B=2, N=2048, D=256, H=8 (dh=32), M=2, NN=2048, K=8. All fp32.

#define Bc   2
#define Nn   2048
#define Dd   256
#define Hh   8
#define DHh  32
#define Mm   2
#define Ll   4096   // M*NN (kv sequence length)

typedef __attribute__((ext_vector_type(2))) float v2f;
typedef __attribute__((ext_vector_type(8))) float v8f;

// ---------------------------------------------------------------------------
// Kernel 1: KNN edge features. One wave per query row.
// out[row] = feat[row] - mean_{t<8} feat[knn_t(row)]
// remap==1: sets are [M,B] flat; output set written at (b*M + m) (concat order).
// ---------------------------------------------------------------------------
__global__ void ea_knn_edge(const float* __restrict__ pts,
                            const float* __restrict__ feats,
                            float* __restrict__ out,
                            int npts, int nsets, int remap)
{
    const int lane = threadIdx.x & 31;
    const int wave = threadIdx.x >> 5;
    const int rowg = blockIdx.x * 8 + wave;
    if (rowg >= nsets * npts) return;
    const int s = rowg / npts;
    const int n = rowg - s * npts;
    int os = s;
    if (remap) { int m = s / Bc; int b = s - m * Bc; os = b * Mm + m; }

    const float* P = pts   + (size_t)s  * npts * 3;
    const float* F = feats + (size_t)s  * npts * Dd;
    float*    Orow = out   + ((size_t)os * npts + n) * Dd;

    const float px = P[n*3+0], py = P[n*3+1], pz = P[n*3+2];

    // per-lane sorted top-8 of packed (dist_bits, idx) keys (ascending)
    unsigned long long key[8];
    #pragma unroll
    for (int t = 0; t < 8; ++t) key[t] = ~0ULL;

    for (int j = lane; j < npts; j += 32) {
        float dx = px - P[j*3+0];
        float dy = py - P[j*3+1];
        float dz = pz - P[j*3+2];
        float d2 = dx*dx + dy*dy + dz*dz;   // >= 0 -> bit pattern is monotone
        unsigned long long k =
            ((unsigned long long)__float_as_uint(d2) << 32) | (unsigned)j;
        if (k < key[7]) {
            key[7] = k;
            #pragma unroll
            for (int t = 7; t > 0; --t) {
                if (key[t] < key[t-1]) {
                    unsigned long long tmp = key[t]; key[t] = key[t-1]; key[t-1] = tmp;
                }
            }
        }
    }

    // wave-wide merge: 8 rounds of 32-lane min-reduction; winner pops its head.
    int head = 0;
    int sel[8];
    #pragma unroll
    for (int t = 0; t < 8; ++t) {
        unsigned long long cand = (head < 8) ? key[head] : ~0ULL;
        unsigned long long mn = cand;
        #pragma unroll
        for (int off = 16; off >= 1; off >>= 1) {
            unsigned long long o = __shfl_xor(mn, off);
            mn = (o < mn) ? o : mn;
        }
        if (cand == mn && head < 8) head++;   // keys are unique (idx embedded)
        sel[t] = (int)(mn & 0xFFFFFFFFu);
    }

    // gather 8 neighbor feature rows, mean, subtract
    for (int c = lane; c < Dd; c += 32) {
        float ssum = 0.f;
        #pragma unroll
        for (int t = 0; t < 8; ++t) ssum += F[(size_t)sel[t] * Dd + c];
        Orow[c] = F[(size_t)n * Dd + c] - ssum * 0.125f;
    }
}

// ---------------------------------------------------------------------------
// Kernel 2: OUT[r][o] = sum_c A[r][c]*W[o][c] + bias[o]  (fp32 WMMA 16x16x4)
// One wave per 16x16 output tile; K-loop of 64 over c.
// ---------------------------------------------------------------------------
__global__ void ea_gemm256(const float* __restrict__ A, const float* __restrict__ W,
                           const float* __restrict__ bias, float* __restrict__ OUT,
                           int R)
{
    const int lane = threadIdx.x & 31;
    const int wave = threadIdx.x >> 5;
    const int tile = blockIdx.x * 8 + wave;
    const int total = (R / 16) * (Dd / 16);
    if (tile >= total) return;
    const int tm = tile >> 4, tn = tile & 15;
    const int col = lane & 15, half = lane >> 4, klo = half * 2;

    const float* Ap = A + (size_t)(tm * 16 + col) * Dd;   // A-frag rows
    const float* Wp = W + (size_t)(tn * 16 + col) * Dd;   // B-frag = W^T cols

    v8f acc = {};
    #pragma unroll 4
    for (int k = 0; k < Dd; k += 4) {
        v2f a = *(const v2f*)(Ap + k + klo);
        v2f b = *(const v2f*)(Wp + k + klo);
        acc = __builtin_amdgcn_wmma_f32_16x16x4_f32(false, a, false, b,
                                                    (short)0, acc, false, false);
    }
    const float bv = bias[tn * 16 + col];
    #pragma unroll
    for (int r = 0; r < 8; ++r) {
        int row = tm * 16 + r + 8 * half;
        OUT[(size_t)row * Dd + tn * 16 + col] = acc[r] + bv;
    }
}

// ---------------------------------------------------------------------------
// Kernel 3: flash attention, one wave per (b, h, 16-row q tile). dh = 32.
// ---------------------------------------------------------------------------
__global__ void ea_attn(const float* __restrict__ Q, const float* __restrict__ Km,
                        const float* __restrict__ Vm, float* __restrict__ O)
{
    __shared__ float plds[8 * 256];          // one 16x16 P tile per wave
    const int lane = threadIdx.x & 31;
    const int wave = threadIdx.x >> 5;
    const int tile = blockIdx.x * 8 + wave;  // grid sized exactly: no guard
    const int qt = tile & 127;
    const int h  = (tile >> 7) & 7;
    const int b  = tile >> 10;
    const int col = lane & 15, half = lane >> 4, klo = half * 2;
    float* pw = plds + wave * 256;

    // preload Q A-frags for this q-tile (dh=32 -> 8 k-steps of 4)
    const float* Qrow = Q + (size_t)(b * Nn + qt * 16 + col) * Dd + h * DHh;
    v2f qf[8];
    #pragma unroll
    for (int kk = 0; kk < 8; ++kk) qf[kk] = *(const v2f*)(Qrow + kk * 4 + klo);

    float mrow[8], lrow[8];
    v8f acc0 = {}, acc1 = {};
    #pragma unroll
    for (int r = 0; r < 8; ++r) { mrow[r] = -__builtin_inff(); lrow[r] = 0.f; }

    const float* Kb = Km + (size_t)b * Ll * Dd + h * DHh;
    const float* Vb = Vm + (size_t)b * Ll * Dd + h * DHh;
    const float scale = 0.17677669529663687f;   // 1/sqrt(32)

    for (int kt = 0; kt < Ll / 16; ++kt) {
        if (kt + 1 < Ll / 16)   // CDNA5 global_prefetch_b8 of next K tile
            __builtin_prefetch(Kb + (size_t)((kt + 1) * 16 + col) * Dd, 0, 1);

        // S = Qh * Kh^T  (16x16), 8 WMMAs
        v8f s = {};
        const float* Kp = Kb + (size_t)(kt * 16 + col) * Dd;
        #pragma unroll
        for (int kk = 0; kk < 8; ++kk) {
            v2f bf = *(const v2f*)(Kp + kk * 4 + klo);
            s = __builtin_amdgcn_wmma_f32_16x16x4_f32(false, qf[kk], false, bf,
                                                      (short)0, s, false, false);
        }

        // online softmax: per-row (16 lanes within each half) reductions
        float p[8];
        #pragma unroll
        for (int r = 0; r < 8; ++r) {
            float v = s[r] * scale;
            float mx = v;
            #pragma unroll
            for (int off = 8; off >= 1; off >>= 1) mx = fmaxf(mx, __shfl_xor(mx, off));
            float mnew = fmaxf(mrow[r], mx);
            float pe = __expf(v - mnew);
            p[r] = pe;
            float sum = pe;
            #pragma unroll
            for (int off = 8; off >= 1; off >>= 1) sum += __shfl_xor(sum, off);
            float corr = __expf(mrow[r] - mnew);
            lrow[r] = lrow[r] * corr + sum;
            acc0[r] *= corr;
            acc1[r] *= corr;
            mrow[r] = mnew;
        }

        // C-layout -> A-layout via per-wave LDS tile
        #pragma unroll
        for (int r = 0; r < 8; ++r) pw[(r + 8 * half) * 16 + col] = p[r];
        __syncthreads();

        // O += P(16x16) * Vh(16x32): two 16x16 D tiles, 4 k-steps each
        #pragma unroll
        for (int kk2 = 0; kk2 < 4; ++kk2) {
            v2f af;
            af.x = pw[col * 16 + kk2 * 4 + klo];
            af.y = pw[col * 16 + kk2 * 4 + klo + 1];
            const float* Vp0 = Vb + (size_t)(kt * 16 + kk2 * 4 + klo) * Dd;
            const float* Vp1 = Vp0 + Dd;
            v2f b0; b0.x = Vp0[col];      b0.y = Vp1[col];
            v2f b1; b1.x = Vp0[16 + col]; b1.y = Vp1[16 + col];
            acc0 = __builtin_amdgcn_wmma_f32_16x16x4_f32(false, af, false, b0,
                                                         (short)0, acc0, false, false);
            acc1 = __builtin_amdgcn_wmma_f32_16x16x4_f32(false, af, false, b1,
                                                         (short)0, acc1, false, false);
        }
        __syncthreads();
    }

    #pragma unroll
    for (int r = 0; r < 8; ++r) {
        float inv = 1.0f / lrow[r];
        int row = b * Nn + qt * 16 + r + 8 * half;
        float* Op = O + (size_t)row * Dd + h * DHh;
        Op[col]      = acc0[r] * inv;
        Op[16 + col] = acc1[r] * inv;
    }
}

// ---------------------------------------------------------------------------
// Kernel 4: out-proj + fused spatial projection epilogue.
// OUT[r][o] = sum_c A[r][c]*W[o][c] + bias[o] + pts[r]·sw[o] + sb[o]
// ---------------------------------------------------------------------------
__global__ void ea_gemm_out(const float* __restrict__ A, const float* __restrict__ W,
                            const float* __restrict__ bias,
                            const float* __restrict__ pts,
                            const float* __restrict__ sw, const float* __restrict__ sb,
                            float* __restrict__ OUT, int R)
{
    const int lane = threadIdx.x & 31;
    const int wave = threadIdx.x >> 5;
    const int tile = blockIdx.x * 8 + wave;
    const int total = (R / 16) * (Dd / 16);
    if (tile >= total) return;
    const int tm = tile >> 4, tn = tile & 15;
    const int col = lane & 15, half = lane >> 4, klo = half * 2;
    const int colg = tn * 16 + col;

    const float* Ap = A + (size_t)(tm * 16 + col) * Dd;
    const float* Wp = W + (size_t)colg * Dd;

    v8f acc = {};
    #pragma unroll 4
    for (int k = 0; k < Dd; k += 4) {
        v2f a = *(const v2f*)(Ap + k + klo);
        v2f b = *(const v2f*)(Wp + k + klo);
        acc = __builtin_amdgcn_wmma_f32_16x16x4_f32(false, a, false, b,
                                                    (short)0, acc, false, false);
    }
    const float s0 = sw[colg*3+0], s1 = sw[colg*3+1], s2 = sw[colg*3+2];
    const float add = bias[colg] + sb[colg];
    #pragma unroll
    for (int r = 0; r < 8; ++r) {
        int row = tm * 16 + r + 8 * half;
        float px = pts[row*3+0], py = pts[row*3+1], pz = pts[row*3+2];
        OUT[(size_t)row * Dd + colg] = acc[r] + add + px*s0 + py*s1 + pz*s2;
    }
}

// ---------------------------------------------------------------------------
extern "C" void kernel_launch(void* const* d_in, const int* in_sizes, int n_in,
                              void* d_out, int out_size, void* d_ws, size_t ws_size,
                              hipStream_t stream)
{
    const float* cur_pts  = (const float*)d_in[0];  // [2,2048,3]
    const float* cur_feat = (const float*)d_in[1];  // [2,2048,256]
    const float* nbr_pts  = (const float*)d_in[2];  // [2,2,2048,3]
    const float* nbr_feat = (const float*)d_in[3];  // [2,2,2048,256]
    const float* in_w     = (const float*)d_in[4];  // [768,256]
    const float* in_b     = (const float*)d_in[5];  // [768]
    const float* out_w    = (const float*)d_in[6];  // [256,256]
    const float* out_b    = (const float*)d_in[7];  // [256]
    const float* sp_w     = (const float*)d_in[8];  // [256,3]
    const float* sp_b     = (const float*)d_in[9];  // [256]
    float* out = (float*)d_out;                     // [2,2048,256]

    // workspace layout (floats): ~36 MB total
    float* ws   = (float*)d_ws;
    float* curE = ws;                        // 4096*256
    float* nbrE = curE + 4096 * 256;         // 8192*256  as [B, M*NN, D]
    float* Qb   = nbrE + 8192 * 256;         // 4096*256
    float* Kb   = Qb   + 4096 * 256;         // 8192*256
    float* Vb   = Kb   + 8192 * 256;         // 8192*256
    float* Ob   = Vb   + 8192 * 256;         // 4096*256

    const dim3 blk(256);

    // 1) KNN edge features
    ea_knn_edge<<<dim3(4096 / 8), blk, 0, stream>>>(cur_pts, cur_feat, curE,
                                                    Nn, Bc, 0);
    ea_knn_edge<<<dim3(8192 / 8), blk, 0, stream>>>(nbr_pts, nbr_feat, nbrE,
                                                    Nn, Mm * Bc, 1);

    // 2) QKV projections (WMMA f32 16x16x4)
    ea_gemm256<<<dim3(4096 / 8), blk, 0, stream>>>(curE, in_w,           in_b,       Qb, 4096);
    ea_gemm256<<<dim3(8192 / 8), blk, 0, stream>>>(nbrE, in_w + 256*256, in_b + 256, Kb, 8192);
    ea_gemm256<<<dim3(8192 / 8), blk, 0, stream>>>(nbrE, in_w + 512*256, in_b + 512, Vb, 8192);

    // 3) flash attention: B*H*(N/16) = 2048 wave-tiles, 8 waves/block
    ea_attn<<<dim3(2048 / 8), blk, 0, stream>>>(Qb, Kb, Vb, Ob);

    // 4) out-proj + spatial epilogue -> d_out
    ea_gemm_out<<<dim3(4096 / 8), blk, 0, stream>>>(Ob, out_w, out_b,
                                                    cur_pts, sp_w, sp_b,
                                                    out, 4096);
}